// OrdinalRegressionLoss_69655779607154
// MI455X (gfx1250) — compile-verified
//
#include <hip/hip_runtime.h>
#include <stdint.h>

#define K            256
#define ROWS_PER_TILE 16
#define BLOCK        256
#define MAX_BLOCKS   8192
#define CLAMP_MIN    (-10.0f)

// Tile = 16 rows x 256 f32 = 16 KiB. Each thread moves 4 x 16B per fill.
#define TILE_FLOATS  (ROWS_PER_TILE * K)          // 4096 floats
#define FILLS_PER_TILE 4                          // 4 x (256 thr x 16 B) = 16 KiB

typedef int v4i __attribute__((ext_vector_type(4)));
typedef __attribute__((address_space(1))) v4i as1_v4i;
typedef __attribute__((address_space(3))) v4i as3_v4i;

// ---------------- async global -> LDS (CDNA5 path, ASYNCcnt) ----------------
__device__ __forceinline__ void async_b128(const float* g, float* l /*shared, generic ptr*/) {
#if __has_builtin(__builtin_amdgcn_global_load_async_to_lds_b128)
    __builtin_amdgcn_global_load_async_to_lds_b128(
        (as1_v4i*)(float*)g, (as3_v4i*)l, 0, 0);
#else
    as3_v4i* lp = (as3_v4i*)l;
    as1_v4i* gp = (as1_v4i*)(float*)g;
    asm volatile("global_load_async_to_lds_b128 %0, %1, off"
                 :: "v"(lp), "v"(gp) : "memory");
#endif
}

template <int N>
__device__ __forceinline__ void wait_asynccnt() {
#if __has_builtin(__builtin_amdgcn_s_wait_asynccnt)
    __builtin_amdgcn_s_wait_asynccnt(N);
#else
    asm volatile("s_wait_asynccnt %0" :: "i"(N) : "memory");
#endif
}

__device__ __forceinline__ void issue_fill(const float* __restrict__ gbase,
                                           float* lbase, int tid) {
#pragma unroll
    for (int i = 0; i < FILLS_PER_TILE; ++i) {
        int foff = (tid + i * BLOCK) * 4;   // float offset, 16B per thread-chunk
        async_b128(gbase + foff, lbase + foff);
    }
}

// ---------------- element contribution ----------------
__device__ __forceinline__ float contrib(float l, int j, int t, float gate) {
    float below = fmaxf(l, CLAMP_MIN);           // term1 piece
    float above = fmaxf(-l, CLAMP_MIN) * gate;   // term2 piece, zeroed if t==K-1
    return (j < t) ? below : above;
}

// ---------------- main streaming kernel ----------------
__global__ void __launch_bounds__(BLOCK)
ord_loss_partial(const float* __restrict__ logits,
                 const int*   __restrict__ targets_i32,  // int64 viewed as int pairs
                 float* __restrict__ partial,
                 int ntiles) {
    __shared__ float smem[2][TILE_FLOATS];
    __shared__ float red[BLOCK / 32];

    const int tid    = threadIdx.x;
    const int bid    = blockIdx.x;
    const int stride = gridDim.x;

    float acc = 0.0f;

    // Prologue: prime both buffers.
    if (bid < ntiles)
        issue_fill(logits + (size_t)bid * TILE_FLOATS, &smem[0][0], tid);
    if (bid + stride < ntiles)
        issue_fill(logits + (size_t)(bid + stride) * TILE_FLOATS, &smem[1][0], tid);

    int buf = 0;
    for (int tile = bid; tile < ntiles; tile += stride) {
        // In flight: fill(tile) [+ fill(tile+stride) if it exists].
        if (tile + stride < ntiles) wait_asynccnt<FILLS_PER_TILE>();
        else                        wait_asynccnt<0>();
        __syncthreads();   // everyone's portion of buf is in LDS

        const int row0 = tile * ROWS_PER_TILE;
#pragma unroll
        for (int i = 0; i < 4; ++i) {
            const int c  = tid + i * BLOCK;   // chunk 0..1023
            const int r  = c >> 6;            // row in tile 0..15
            const int j0 = (c & 63) << 2;     // start column
            const int t  = targets_i32[(size_t)(row0 + r) * 2]; // low dword of int64
            const float gate = (t < K - 1) ? 1.0f : 0.0f;
            const float4 v = *(const float4*)&smem[buf][r * K + j0];
            acc += contrib(v.x, j0 + 0, t, gate);
            acc += contrib(v.y, j0 + 1, t, gate);
            acc += contrib(v.z, j0 + 2, t, gate);
            acc += contrib(v.w, j0 + 3, t, gate);
        }
        __syncthreads();   // all reads of buf done — safe to overwrite

        const int tnext = tile + 2 * stride;
        if (tnext < ntiles)
            issue_fill(logits + (size_t)tnext * TILE_FLOATS, &smem[buf][0], tid);
        buf ^= 1;
    }

    // wave32 reduction, then cross-wave via LDS
#pragma unroll
    for (int o = 16; o > 0; o >>= 1) acc += __shfl_xor(acc, o, 32);
    if ((tid & 31) == 0) red[tid >> 5] = acc;
    __syncthreads();
    if (tid == 0) {
        float s = 0.0f;
#pragma unroll
        for (int w = 0; w < BLOCK / 32; ++w) s += red[w];
        partial[bid] = s;
    }
}

// ---------------- final reduction ----------------
__global__ void __launch_bounds__(BLOCK)
ord_loss_final(const float* __restrict__ partial, int n,
               float* __restrict__ out, float inv_batch) {
    __shared__ float red[BLOCK / 32];
    float s = 0.0f;
    for (int i = threadIdx.x; i < n; i += BLOCK) s += partial[i];
#pragma unroll
    for (int o = 16; o > 0; o >>= 1) s += __shfl_xor(s, o, 32);
    if ((threadIdx.x & 31) == 0) red[threadIdx.x >> 5] = s;
    __syncthreads();
    if (threadIdx.x == 0) {
        float t = 0.0f;
#pragma unroll
        for (int w = 0; w < BLOCK / 32; ++w) t += red[w];
        out[0] = t * inv_batch;
    }
}

extern "C" void kernel_launch(void* const* d_in, const int* in_sizes, int n_in,
                              void* d_out, int out_size, void* d_ws, size_t ws_size,
                              hipStream_t stream) {
    const float* logits  = (const float*)d_in[0];
    const int*   tgt_i32 = (const int*)d_in[1];    // int64 targets, read low dwords
    float*       out     = (float*)d_out;
    float*       partial = (float*)d_ws;

    const int rows   = in_sizes[1];                // batch = 524288
    const int ntiles = rows / ROWS_PER_TILE;       // batch divisible by 16

    int grid = ntiles < MAX_BLOCKS ? ntiles : MAX_BLOCKS;
    size_t need = (size_t)grid * sizeof(float);
    if (need > ws_size && ws_size >= sizeof(float))
        grid = (int)(ws_size / sizeof(float));

    ord_loss_partial<<<grid, BLOCK, 0, stream>>>(logits, tgt_i32, partial, ntiles);
    ord_loss_final<<<1, BLOCK, 0, stream>>>(partial, grid, out, 1.0f / (float)rows);
}